// CVNonGaussianQuantumLayer_60610578482019
// MI455X (gfx1250) — compile-verified
//
#include <hip/hip_runtime.h>
#include <math.h>

#define NQ 12
#define PADIDX(i) ((i) + ((i) >> 6))
#define PI_F 3.14159265358979323846f

typedef float v2f __attribute__((ext_vector_type(2)));
typedef float v8f __attribute__((ext_vector_type(8)));

struct cpx { float re, im; };
__device__ __forceinline__ cpx cmul(cpx a, cpx b){ return {a.re*b.re - a.im*b.im, a.re*b.im + a.im*b.re}; }
__device__ __forceinline__ cpx cadd(cpx a, cpx b){ return {a.re+b.re, a.im+b.im}; }
__device__ __forceinline__ float sigm(float v){ return 1.0f/(1.0f + expf(-v)); }

// Factor chains building the four fused 16x16 gates per layer.
// pass 0: wires 0-3 (bits 11..8), pass 1: wires 3-6 (bits 8..5),
// pass 2: wires 6-9 (bits 5..2), pass 3: wires 8-11 (bits 3..0).
// type: 0 = identity, 1 = 1q U(wire=idx) on local bit `bit`,
//       2 = CRX(theta idx) target local bit `bit`, control bit = bit+1
__constant__ int c_ftype[4][7] = {
  {1,2,1,2,1,2,1},
  {2,1,2,1,2,1,0},
  {2,1,2,1,2,1,0},
  {2,1,2,1,0,0,0}};
__constant__ int c_fbit[4][7] = {
  {3,2,2,1,1,0,0},
  {2,2,1,1,0,0,0},
  {2,2,1,1,0,0,0},
  {1,1,0,0,0,0,0}};
__constant__ int c_fidx[4][7] = {
  {0,0,1,1,2,2,3},
  {3,4,4,5,5,6,0},
  {6,7,7,8,8,9,0},
  {9,10,10,11,0,0,0}};

__global__ __launch_bounds__(256)
void qsim12_wmma_kernel(const float* __restrict__ x,
                        const float* __restrict__ rot,
                        const float* __restrict__ cxs,
                        const float* __restrict__ tg,
                        float* __restrict__ out)
{
    // State (4096 complex) padded every 64 floats to dodge bank conflicts.
    __shared__ float sre[4160];
    __shared__ float sim_[4160];
    // 8 fused 16x16 complex gates: [layer*4 + pass]
    __shared__ float Wre[8][16][16];
    __shared__ float Wim[8][16][16];
    // composed 1q gates U = Phase*Rz*Ry*Rx : [layer][wire][4 entries]
    __shared__ float Ure[2][12][4];
    __shared__ float Uim[2][12][4];
    // CRX half-angle cos/sin: [layer][theta idx]
    __shared__ float Cc[2][11];
    __shared__ float Cs[2][11];
    __shared__ float cw[12], sw[12];
    __shared__ float red[96];

    const int tid    = threadIdx.x;
    const int lane   = tid & 31;
    const int wav    = tid >> 5;
    const int batch  = blockIdx.x & 127;
    const int branch = blockIdx.x >> 7;

    // gfx1250: prefetch tiny parameter arrays (global_prefetch_b8)
    if (tid == 0){
        __builtin_prefetch(rot, 0, 3);
        __builtin_prefetch(cxs, 0, 3);
        __builtin_prefetch(tg,  0, 3);
        __builtin_prefetch(x + batch*NQ, 0, 3);
    }

    // ---------- Phase 1: parameters ----------
    if (tid < NQ){
        float xv = x[batch*NQ + tid];
        float h = 0.5f * PI_F * atan2f(xv, 1.0f);
        cw[tid] = cosf(h); sw[tid] = sinf(h);
    }
    if (tid >= 32 && tid < 56){
        int idx = tid - 32; int L = idx / 12; int w = idx % 12;
        int rbase = ((branch*2 + L)*12 + w)*3;
        float hx = 0.5f*rot[rbase+0], hy = 0.5f*rot[rbase+1], hz = 0.5f*rot[rbase+2];
        float cx_ = cosf(hx), sx_ = sinf(hx);
        float cy_ = cosf(hy), sy_ = sinf(hy);
        // M = Ry * Rx
        cpx M00 = { cy_*cx_,  sy_*sx_ };
        cpx M01 = {-sy_*cx_, -cy_*sx_ };
        cpx M10 = { sy_*cx_, -cy_*sx_ };
        cpx M11 = { cy_*cx_, -sy_*sx_ };
        cpx e0 = { cosf(hz), -sinf(hz) };        // exp(-i hz)
        cpx e1 = { e0.re, -e0.im };              // exp(+i hz)
        float phi = sigm(tg[(branch*2 + L)*12 + w]) * PI_F;
        cpx ep = { cosf(phi), sinf(phi) };
        cpx r1 = cmul(ep, e1);
        cpx U00 = cmul(e0, M00), U01 = cmul(e0, M01);
        cpx U10 = cmul(r1, M10), U11 = cmul(r1, M11);
        Ure[L][w][0] = U00.re; Uim[L][w][0] = U00.im;
        Ure[L][w][1] = U01.re; Uim[L][w][1] = U01.im;
        Ure[L][w][2] = U10.re; Uim[L][w][2] = U10.im;
        Ure[L][w][3] = U11.re; Uim[L][w][3] = U11.im;
    }
    if (tid >= 64 && tid < 86){
        int idx = tid - 64; int L = idx / 11; int w = idx % 11;
        float th = sigm(cxs[(branch*2 + L)*11 + w]) * PI_F;
        float h = 0.5f*th;
        Cc[L][w] = cosf(h); Cs[L][w] = sinf(h);
    }
    __syncthreads();

    // ---------- Phase 2a: initial product state ----------
    {
        float prefix = 1.0f;
        #pragma unroll
        for (int b = 4; b <= 11; ++b){
            int bit = (tid >> (b-4)) & 1;
            int w = 11 - b;
            prefix *= bit ? sw[w] : cw[w];
        }
        #pragma unroll
        for (int k = 0; k < 16; ++k){
            float v = prefix;
            #pragma unroll
            for (int b = 0; b < 4; ++b){
                int bit = (k >> b) & 1;
                int w = 11 - b;
                v *= bit ? sw[w] : cw[w];
            }
            int i = tid*16 + k;
            sre[PADIDX(i)] = v;
            sim_[PADIDX(i)] = 0.0f;
        }
    }

    // ---------- Phase 2b: build fused 16x16 gates (one wave per matrix) ----------
    {
        const int mm = wav;            // matrix 0..7
        const int Lm = mm >> 2, pm = mm & 3;
        #pragma unroll
        for (int jj = 0; jj < 8; ++jj){
            int e = lane*8 + jj, r = e >> 4, c = e & 15;
            Wre[mm][r][c] = (r == c) ? 1.0f : 0.0f;
            Wim[mm][r][c] = 0.0f;
        }
        __syncthreads();
        for (int s = 0; s < 7; ++s){
            int ft = c_ftype[pm][s], fb = c_fbit[pm][s], fi = c_fidx[pm][s];
            float nr[8], ni[8];
            #pragma unroll
            for (int jj = 0; jj < 8; ++jj){
                int e = lane*8 + jj, r = e >> 4, c = e & 15;
                cpx nv;
                if (ft == 1){
                    int rb = (r >> fb) & 1;
                    int r0 = r & ~(1 << fb), r1 = r | (1 << fb);
                    cpx u0 = {Ure[Lm][fi][rb*2+0], Uim[Lm][fi][rb*2+0]};
                    cpx u1 = {Ure[Lm][fi][rb*2+1], Uim[Lm][fi][rb*2+1]};
                    cpx o0 = {Wre[mm][r0][c], Wim[mm][r0][c]};
                    cpx o1 = {Wre[mm][r1][c], Wim[mm][r1][c]};
                    nv = cadd(cmul(u0,o0), cmul(u1,o1));
                } else if (ft == 2){
                    int cb = fb + 1;
                    if (((r >> cb) & 1) == 0){
                        nv = {Wre[mm][r][c], Wim[mm][r][c]};
                    } else {
                        int a = (r >> fb) & 1;
                        int r0 = r & ~(1 << fb), r1 = r | (1 << fb);
                        float ccv = Cc[Lm][fi], ssv = Cs[Lm][fi];
                        cpx d = {ccv, 0.0f}, o = {0.0f, -ssv};
                        cpx m0 = a ? o : d;
                        cpx m1 = a ? d : o;
                        cpx o0 = {Wre[mm][r0][c], Wim[mm][r0][c]};
                        cpx o1 = {Wre[mm][r1][c], Wim[mm][r1][c]};
                        nv = cadd(cmul(m0,o0), cmul(m1,o1));
                    }
                } else {
                    nv = {Wre[mm][r][c], Wim[mm][r][c]};
                }
                nr[jj] = nv.re; ni[jj] = nv.im;
            }
            __syncthreads();
            #pragma unroll
            for (int jj = 0; jj < 8; ++jj){
                int e = lane*8 + jj, r = e >> 4, c = e & 15;
                Wre[mm][r][c] = nr[jj];
                Wim[mm][r][c] = ni[jj];
            }
            __syncthreads();
        }
    }

    // ---------- Phase 3: evolution, 4 WMMA passes per layer ----------
    // Pass p: state viewed as 256 column-groups of 16 amplitudes with
    // element addr = (col>>lcs)<<(lcs+4) | k<<lcs | (col & (2^lcs -1))
    const int lcs_tab[4] = {8, 5, 2, 0};
    const int n  = lane & 15;      // B/D column within tile, A row
    const int kh = lane >> 4;      // operand half (K/M split across lane halves)

    for (int L = 0; L < 2; ++L){
        for (int p = 0; p < 4; ++p){
            const int mi = L*4 + p;
            const int lcs = lcs_tab[p];
            // A operands (gate matrix), hoisted: chunks j cover K = 4j..4j+3
            v2f are[4], aim[4], ani[4];
            #pragma unroll
            for (int j = 0; j < 4; ++j){
                int k0 = 4*j + 2*kh;
                v2f tr = { Wre[mi][n][k0], Wre[mi][n][k0+1] };
                v2f ti = { Wim[mi][n][k0], Wim[mi][n][k0+1] };
                are[j] = tr; aim[j] = ti; ani[j] = -ti;
            }
            #pragma unroll
            for (int t2 = 0; t2 < 2; ++t2){
                int col  = (wav*2 + t2)*16 + n;
                int hi   = col >> lcs;
                int lo   = col - (hi << lcs);
                int base = (hi << (lcs + 4)) + lo;
                v2f bre[4], bim[4];
                #pragma unroll
                for (int j = 0; j < 4; ++j){
                    int k0 = 4*j + 2*kh;
                    int a0 = base + (k0 << lcs);
                    int a1 = base + ((k0+1) << lcs);
                    v2f tr = { sre[PADIDX(a0)], sre[PADIDX(a1)] };
                    v2f ti = { sim_[PADIDX(a0)], sim_[PADIDX(a1)] };
                    bre[j] = tr; bim[j] = ti;
                }
                v8f cre = {0,0,0,0,0,0,0,0};
                v8f cim = {0,0,0,0,0,0,0,0};
                #pragma unroll
                for (int j = 0; j < 4; ++j)
                    cre = __builtin_amdgcn_wmma_f32_16x16x4_f32(false, are[j], false, bre[j], (short)0, cre, false, false);
                #pragma unroll
                for (int j = 0; j < 4; ++j)
                    cre = __builtin_amdgcn_wmma_f32_16x16x4_f32(false, ani[j], false, bim[j], (short)0, cre, false, false);
                #pragma unroll
                for (int j = 0; j < 4; ++j)
                    cim = __builtin_amdgcn_wmma_f32_16x16x4_f32(false, are[j], false, bim[j], (short)0, cim, false, false);
                #pragma unroll
                for (int j = 0; j < 4; ++j)
                    cim = __builtin_amdgcn_wmma_f32_16x16x4_f32(false, aim[j], false, bre[j], (short)0, cim, false, false);
                // scatter D (rows m = v + 16*? -> v + kh*8) back in place
                #pragma unroll
                for (int v = 0; v < 8; ++v){
                    int m  = v + kh*8;
                    int am = base + (m << lcs);
                    sre[PADIDX(am)] = cre[v];
                    sim_[PADIDX(am)] = cim[v];
                }
            }
            __syncthreads();
        }
        if (L == 0){
            // CNOT ladder == Gray-code permutation psi'[i] = psi[i ^ (i>>1)]
            float tre[16], tim2[16];
            #pragma unroll
            for (int k = 0; k < 16; ++k){
                int i = tid*16 + k;
                int g = i ^ (i >> 1);
                tre[k]  = sre[PADIDX(g)];
                tim2[k] = sim_[PADIDX(g)];
            }
            __syncthreads();
            #pragma unroll
            for (int k = 0; k < 16; ++k){
                int i = tid*16 + k;
                sre[PADIDX(i)]  = tre[k];
                sim_[PADIDX(i)] = tim2[k];
            }
            __syncthreads();
        }
    }

    // ---------- Phase 4: expectations (layer-1 Gray perm folded into reads) ----------
    float part[12];
    if (branch == 0){
        // <X_w> = 2 * sum_{i: bit_p(i)=0} Re(conj(a_i) * a_{i|2^p})
        for (int w = 0; w < 12; ++w){
            int pb = 11 - w; int msk = 1 << pb;
            float acc = 0.0f;
            #pragma unroll
            for (int qq = 0; qq < 8; ++qq){
                int q  = tid*8 + qq;
                int i0 = ((q >> pb) << (pb+1)) | (q & (msk-1));
                int i1 = i0 | msk;
                int g0 = i0 ^ (i0 >> 1);
                int g1 = i1 ^ (i1 >> 1);
                acc += sre[PADIDX(g0)]*sre[PADIDX(g1)] + sim_[PADIDX(g0)]*sim_[PADIDX(g1)];
            }
            part[w] = 2.0f * acc;
        }
    } else {
        // <Z_w> = sum_i (+-)|a_i|^2
        float mods[16];
        #pragma unroll
        for (int k = 0; k < 16; ++k){
            int i = tid*16 + k;
            int g = i ^ (i >> 1);
            float a = sre[PADIDX(g)], b = sim_[PADIDX(g)];
            mods[k] = a*a + b*b;
        }
        for (int w = 0; w < 12; ++w){
            int pb = 11 - w;
            float acc = 0.0f;
            #pragma unroll
            for (int k = 0; k < 16; ++k){
                int i = tid*16 + k;
                acc += ((i >> pb) & 1) ? -mods[k] : mods[k];
            }
            part[w] = acc;
        }
    }
    // wave32 reduce then cross-wave via LDS (deterministic, single writer)
    for (int w = 0; w < 12; ++w){
        float v = part[w];
        v += __shfl_xor(v, 16, 32);
        v += __shfl_xor(v,  8, 32);
        v += __shfl_xor(v,  4, 32);
        v += __shfl_xor(v,  2, 32);
        v += __shfl_xor(v,  1, 32);
        if (lane == 0) red[w*8 + wav] = v;
    }
    __syncthreads();
    if (tid < 12){
        float s = 0.0f;
        #pragma unroll
        for (int k = 0; k < 8; ++k) s += red[tid*8 + k];
        out[batch*24 + 2*tid + branch] = s;
    }
}

extern "C" void kernel_launch(void* const* d_in, const int* in_sizes, int n_in,
                              void* d_out, int out_size, void* d_ws, size_t ws_size,
                              hipStream_t stream) {
    const float* x   = (const float*)d_in[0];   // (128, 12)
    const float* rot = (const float*)d_in[1];   // (2, 2, 12, 3)
    const float* cxs = (const float*)d_in[2];   // (2, 2, 11)
    const float* tg  = (const float*)d_in[3];   // (2, 2, 12)
    float* out = (float*)d_out;                 // (128, 24)
    (void)in_sizes; (void)n_in; (void)out_size; (void)d_ws; (void)ws_size;
    qsim12_wmma_kernel<<<dim3(256), dim3(256), 0, stream>>>(x, rot, cxs, tg, out);
}